// BatchLossFunction_38534446579748
// MI455X (gfx1250) — compile-verified
//
#include <hip/hip_runtime.h>
#include <math.h>

typedef __attribute__((ext_vector_type(2))) float v2f;
typedef __attribute__((ext_vector_type(4))) float v4f;
typedef __attribute__((ext_vector_type(8))) float v8f;

#define GAMMA_C   3.4f
#define EPS_C     1e-8f
#define B_SIZE    2048
#define P_TOK     197
#define P_USE     196
#define D_DIM     768
#define KC        64            // K-chunk staged in LDS per wave
#define NCHUNK    (D_DIM / KC)  // 12
#define ROW_STRIDE 68           // padded LDS row stride (floats): 64-bank conflict-free
#define WAVES     13
#define THREADS   (WAVES * 32)

__global__ void zero_out_kernel(float* out) { out[0] = 0.0f; }

__global__ __launch_bounds__(THREADS)
void batch_loss_kernel(const float* __restrict__ pt,   // (B, 197, 768)
                       const float* __restrict__ ot,   // (B, 768)
                       const float* __restrict__ gt,   // (B, 196)
                       float* __restrict__ out)        // scalar
{
    __shared__ float s_ot[D_DIM];
    __shared__ float s_tile[WAVES * 16 * ROW_STRIDE];
    __shared__ float s_partial[WAVES];

    const int b    = blockIdx.x;
    const int tid  = threadIdx.x;
    const int wave = tid >> 5;
    const int lane = tid & 31;

    // Stage out_text[b] into LDS (coalesced).
    for (int i = tid; i < D_DIM; i += THREADS)
        s_ot[i] = ot[(size_t)b * D_DIM + i];
    __syncthreads();

    // ||out_text||^2, computed per wave from LDS.
    float ot2 = 0.0f;
    for (int i = lane; i < D_DIM; i += 32) {
        float v = s_ot[i];
        ot2 += v * v;
    }
    #pragma unroll
    for (int off = 16; off > 0; off >>= 1)
        ot2 += __shfl_xor(ot2, off, 32);

    float* tile    = &s_tile[wave * 16 * ROW_STRIDE];
    const int p0   = wave * 16;
    const int m    = lane & 15;          // row for A-operand reads
    const int koff = (lane >> 4) << 1;   // k offset: 0 (lanes 0-15) / 2 (lanes 16-31)
    const int rsub = lane >> 4;          // row-pair half for staging
    const int cidx = (lane & 15) * 4;    // column (floats) for b128 staging

    // Per-chunk global load: 8 x b128 per lane; lanes split row pairs.
    // 32 lanes x 16B = 512B per instruction, fully coalesced per row.
    auto load_chunk = [&](v4f (&buf)[8], int kc) {
        #pragma unroll
        for (int r2 = 0; r2 < 8; ++r2) {
            const int r = r2 * 2 + rsub;
            const int p = p0 + r;
            v4f v = {0.0f, 0.0f, 0.0f, 0.0f};
            if (p < P_USE) {                 // tail rows zero-filled (keeps WMMA EXEC full)
                const float* src =
                    pt + ((size_t)(b * P_TOK + 1 + p) * D_DIM + kc) + cidx;
                v = *(const v4f*)src;
            }
            buf[r2] = v;
        }
    };
    auto store_chunk = [&](const v4f (&buf)[8]) {
        #pragma unroll
        for (int r2 = 0; r2 < 8; ++r2) {
            const int r = r2 * 2 + rsub;
            *(v4f*)(&tile[r * ROW_STRIDE + cidx]) = buf[r2];
        }
    };

    v8f   c  = {};     // 16x16 f32 accumulator (dot products in every column)
    float sq = 0.0f;   // partial row sum-of-squares (this lane's k residues)
    v4f   buf[8];

    // Software pipeline: prefetch chunk k+1 into registers while WMMAs
    // consume chunk k from LDS (hides HBM latency inside each wave).
    load_chunk(buf, 0);
    for (int kci = 0; kci < NCHUNK; ++kci) {
        const int kc = kci * KC;
        store_chunk(buf);
        if (kci + 1 < NCHUNK)
            load_chunk(buf, kc + KC);

        #pragma unroll
        for (int k4 = 0; k4 < KC; k4 += 4) {
            const int kk = k4 + koff;
            v2f a = *(const v2f*)(&tile[m * ROW_STRIDE + kk]);   // A: 16x4 f32 layout
            v2f bv;                                              // B: ot broadcast to 16 cols
            bv.x = s_ot[kc + kk];
            bv.y = s_ot[kc + kk + 1];
            sq += a.x * a.x + a.y * a.y;
            c = __builtin_amdgcn_wmma_f32_16x16x4_f32(false, a, false, bv,
                                                      (short)0, c, false, false);
        }
    }

    // Combine the two k-half partials: lane m holds full ||row m||^2.
    sq += __shfl_xor(sq, 16, 32);

    // C layout: lane 0 holds rows 0-7 in c[0..7]; lane 16 holds rows 8-15.
    float acc = 0.0f;
    const int half = lane >> 4;
    #pragma unroll
    for (int j = 0; j < 8; ++j) {
        const int row   = half * 8 + j;
        const float rn2 = __shfl(sq, row, 32);   // executed by all lanes
        const float dot = c[j];
        if ((lane & 15) == 0) {
            const int p = p0 + row;
            if (p < P_USE) {
                const float ptn  = fmaxf(sqrtf(rn2), EPS_C);
                const float otn  = fmaxf(sqrtf(ot2), EPS_C);
                const float cosv = dot / (ptn * otn);
                const float vec  = 1.0f / (1.0f + __expf(cosv - 1.0f)); // sigmoid(1-cos)
                const float g    = gt[(size_t)b * P_USE + p] * (1.0f / 255.0f);
                const float diff = fabsf(vec - g);
                acc += -__logf(1.0f - diff) * (g * GAMMA_C + 1.0f);
            }
        }
    }

    // Combine lane 0 + lane 16 partials, then block-reduce across waves.
    acc += __shfl_xor(acc, 16, 32);
    if (lane == 0) s_partial[wave] = acc;
    __syncthreads();

    if (tid == 0) {
        float s = 0.0f;
        #pragma unroll
        for (int w = 0; w < WAVES; ++w) s += s_partial[w];
        atomicAdd(out, s * (1.0f / (float)B_SIZE));
    }
}

extern "C" void kernel_launch(void* const* d_in, const int* in_sizes, int n_in,
                              void* d_out, int out_size, void* d_ws, size_t ws_size,
                              hipStream_t stream) {
    const float* pt = (const float*)d_in[0];   // (2048, 197, 768) f32
    const float* ot = (const float*)d_in[1];   // (2048, 768) f32
    const float* gt = (const float*)d_in[2];   // (2048, 14, 14) f32
    float* out = (float*)d_out;                // scalar f32

    zero_out_kernel<<<1, 1, 0, stream>>>(out);
    batch_loss_kernel<<<B_SIZE, THREADS, 0, stream>>>(pt, ot, gt, out);
}